// MemStream_57501022159244
// MI455X (gfx1250) — compile-verified
//
#include <hip/hip_runtime.h>
#include <hip/hip_bf16.h>

typedef __attribute__((ext_vector_type(2))) float v2f;
typedef __attribute__((ext_vector_type(8))) float v8f;

#define B_ROWS   256
#define D_IN     128
#define D_OUT    256
#define MEM_LEN  4096
#define BETA     50.0f

// ---------------------------------------------------------------------------
// Kernel 0: hoist the normalization out of the GEMM.
//   scale[k] = (std[k]==0) ? 0 : 1/std[k]
//   shift[k] = mean[k]*scale[k]
// so new[i,k] = fma(x[i,k], scale[k], -shift[k]) with the zero-std case
// handled implicitly (scale=0 kills both terms). 128 divisions total on the
// whole device instead of 64 per lane in the hot loop.
// ---------------------------------------------------------------------------
__global__ __launch_bounds__(128) void prep_norm_kernel(
    const float* __restrict__ mean, const float* __restrict__ std_,
    float* __restrict__ scale, float* __restrict__ shift) {
  const int k = threadIdx.x;
  const float s = std_[k];
  const float sc = (s == 0.0f) ? 0.0f : 1.0f / s;
  scale[k] = sc;
  shift[k] = mean[k] * sc;
}

// ---------------------------------------------------------------------------
// Kernel 1: enc = tanh(new @ W1 + b1) via native fp32 WMMA.
// Grid: 32 blocks x 256 threads = 256 waves; one 16x16 output tile per wave.
// K loop: 128/4 = 32 x V_WMMA_F32_16X16X4_F32, branch-free (EXEC all-1s).
// A(16x4): lanes 0-15 hold K={0,1}, lanes 16-31 hold K={2,3}, row = lane%16.
// B(4x16): col = lane%16, same K split. D(16x16): VGPR r -> M = r + 8*(lane>=16).
// ---------------------------------------------------------------------------
__global__ __launch_bounds__(256) void enc_gemm_wmma(
    const float* __restrict__ x, const float* __restrict__ scale,
    const float* __restrict__ shift, const float* __restrict__ W1,
    const float* __restrict__ b1, float* __restrict__ enc) {
  const int wave = threadIdx.x >> 5;          // 0..7
  const int lane = threadIdx.x & 31;
  const int tile = blockIdx.x * 8 + wave;     // 0..255
  const int tm = tile >> 4;                   // tile row (M/16)
  const int tn = tile & 15;                   // tile col (N/16)
  const int half = lane >> 4;                 // 0: lanes 0-15, 1: lanes 16-31
  const int l16 = lane & 15;

  const int rowA = tm * 16 + l16;             // A-matrix row for this lane
  const int colB = tn * 16 + l16;             // B-matrix col for this lane
  const float* __restrict__ xrow = x + (size_t)rowA * D_IN;

  v8f c = {};
#pragma unroll
  for (int k0 = 0; k0 < D_IN; k0 += 4) {
    const int ka = k0 + half * 2;             // this lane's contiguous K pair
    const float2 xv = *reinterpret_cast<const float2*>(xrow + ka);
    const float2 sc = *reinterpret_cast<const float2*>(scale + ka);
    const float2 sh = *reinterpret_cast<const float2*>(shift + ka);
    v2f A;
    A[0] = fmaf(xv.x, sc.x, -sh.x);
    A[1] = fmaf(xv.y, sc.y, -sh.y);
    v2f Bf;
    Bf[0] = W1[(size_t)ka * D_OUT + colB];
    Bf[1] = W1[(size_t)(ka + 1) * D_OUT + colB];
    c = __builtin_amdgcn_wmma_f32_16x16x4_f32(
        /*neg_a=*/false, A, /*neg_b=*/false, Bf,
        /*c_mod=*/(short)0, c, /*reuse_a=*/false, /*reuse_b=*/false);
  }

  const int baseRow = tm * 16 + half * 8;
  const int col = tn * 16 + l16;
  const float bias = b1[col];
#pragma unroll
  for (int r = 0; r < 8; ++r) {
    enc[(size_t)(baseRow + r) * D_OUT + col] = tanhf(c[r] + bias);
  }
}

// ---------------------------------------------------------------------------
// Kernel 2: L1 cdist row-min. One block per batch row; enc row staged in LDS;
// each thread streams 16 memory rows with float4 loads (memory is L2-resident:
// 4 MB << 192 MB), prefetching the next row it will touch.
// ---------------------------------------------------------------------------
__global__ __launch_bounds__(256) void cdist_min_kernel(
    const float* __restrict__ enc, const float* __restrict__ memory,
    float* __restrict__ loss) {
  __shared__ float encRow[D_OUT];
  __shared__ float red[256];
  const int b = blockIdx.x;
  const int t = threadIdx.x;

  encRow[t] = enc[(size_t)b * D_OUT + t];
  __syncthreads();

  float best = 3.402823466e38f;
  for (int i = 0; i < MEM_LEN / 256; ++i) {
    const int m = i * 256 + t;
    const float4* __restrict__ mrow =
        reinterpret_cast<const float4*>(memory + (size_t)m * D_OUT);
    if (i + 1 < MEM_LEN / 256) {
      __builtin_prefetch(memory + (size_t)(m + 256) * D_OUT, 0, 0);
    }
    float s = 0.0f;
#pragma unroll 8
    for (int d4 = 0; d4 < D_OUT / 4; ++d4) {
      const float4 mv = mrow[d4];
      const int d = d4 * 4;
      s += fabsf(encRow[d]     - mv.x);
      s += fabsf(encRow[d + 1] - mv.y);
      s += fabsf(encRow[d + 2] - mv.z);
      s += fabsf(encRow[d + 3] - mv.w);
    }
    best = fminf(best, s);
  }

  red[t] = best;
  __syncthreads();
#pragma unroll
  for (int off = 128; off > 0; off >>= 1) {
    if (t < off) red[t] = fminf(red[t], red[t + off]);
    __syncthreads();
  }
  if (t == 0) loss[b] = red[0];
}

// ---------------------------------------------------------------------------
// Kernel 3: predicate + exclusive prefix sum -> scatter positions.
// count starts at MEM_LEN (first pos wraps to 0) and <=256 writes into a
// 4096-deep ring never wrap or collide, so positions are just the exclusive
// scan of the predicate and all scatters are independent.
// ---------------------------------------------------------------------------
__global__ __launch_bounds__(256) void prefix_pos_kernel(
    const float* __restrict__ loss, int* __restrict__ pos,
    int* __restrict__ pred) {
  __shared__ int sp[256];
  const int t = threadIdx.x;
  const float l = loss[t];
  const int p = ((l == l) && (l <= BETA)) ? 1 : 0;  // NaN/Inf fail l<=BETA
  sp[t] = p;
  __syncthreads();
  for (int off = 1; off < 256; off <<= 1) {
    int v = 0;
    if (t >= off) v = sp[t - off];
    __syncthreads();
    sp[t] += v;
    __syncthreads();
  }
  pos[t] = sp[t] - p;  // exclusive scan
  pred[t] = p;
}

// ---------------------------------------------------------------------------
// Kernel 4: parallel scatter of accepted rows into the copied-out buffers.
// ---------------------------------------------------------------------------
__global__ __launch_bounds__(128) void scatter_kernel(
    const float* __restrict__ enc, const float* __restrict__ x,
    const int* __restrict__ pos, const int* __restrict__ pred,
    float* __restrict__ mem_out, float* __restrict__ md_out) {
  const int b = blockIdx.x;
  if (!pred[b]) return;
  const int p = pos[b];
  const int t = threadIdx.x;
  mem_out[(size_t)p * D_OUT + t]       = enc[(size_t)b * D_OUT + t];
  mem_out[(size_t)p * D_OUT + 128 + t] = enc[(size_t)b * D_OUT + 128 + t];
  md_out[(size_t)p * D_IN + t]         = x[(size_t)b * D_IN + t];
}

extern "C" void kernel_launch(void* const* d_in, const int* in_sizes, int n_in,
                              void* d_out, int out_size, void* d_ws, size_t ws_size,
                              hipStream_t stream) {
  const float* x        = (const float*)d_in[0];  // [256,128]
  const float* mean     = (const float*)d_in[1];  // [128]
  const float* std_     = (const float*)d_in[2];  // [128]
  const float* W1       = (const float*)d_in[3];  // [128,256]
  const float* b1       = (const float*)d_in[4];  // [256]
  const float* memory   = (const float*)d_in[5];  // [4096,256]
  const float* mem_data = (const float*)d_in[6];  // [4096,128]

  float* out     = (float*)d_out;
  float* loss    = out;                                    // [256]
  float* mem_out = out + B_ROWS;                           // [4096,256]
  float* md_out  = out + B_ROWS + (size_t)MEM_LEN * D_OUT; // [4096,128]

  float* enc   = (float*)d_ws;                             // [256,256]
  int*   pos   = (int*)((char*)d_ws + (size_t)B_ROWS * D_OUT * sizeof(float));
  int*   pred  = pos + B_ROWS;
  float* scale = (float*)(pred + B_ROWS);                  // [128]
  float* shift = scale + D_IN;                             // [128]

  // 0) fold mean/std into a branch-free scale/shift pair
  prep_norm_kernel<<<1, 128, 0, stream>>>(mean, std_, scale, shift);

  // 1) encoder GEMM (WMMA fp32) + tanh
  enc_gemm_wmma<<<32, 256, 0, stream>>>(x, scale, shift, W1, b1, enc);

  // 2) copy original memory / mem_data into outputs (overwritten by scatter)
  hipMemcpyAsync(mem_out, memory,   (size_t)MEM_LEN * D_OUT * sizeof(float),
                 hipMemcpyDeviceToDevice, stream);
  hipMemcpyAsync(md_out,  mem_data, (size_t)MEM_LEN * D_IN  * sizeof(float),
                 hipMemcpyDeviceToDevice, stream);

  // 3) L1 cdist + row min -> loss
  cdist_min_kernel<<<B_ROWS, 256, 0, stream>>>(enc, memory, loss);

  // 4) predicate prefix scan -> write positions
  prefix_pos_kernel<<<1, 256, 0, stream>>>(loss, pos, pred);

  // 5) scatter accepted rows
  scatter_kernel<<<B_ROWS, 128, 0, stream>>>(enc, x, pos, pred, mem_out, md_out);
}